// QuanvolutionHybrid_65481071405301
// MI455X (gfx1250) — compile-verified
//
#include <hip/hip_runtime.h>
#include <math.h>

#define BSZ 16384
#define NFEAT 784
#define NH 256

typedef __attribute__((ext_vector_type(2))) float v2f;
typedef __attribute__((ext_vector_type(8))) float v8f;

// ---------------------------------------------------------------------------
// Quantum circuit simulation, lane-parallel: one amplitude per lane,
// 2 images per wave32 (lanes 0-15 = image A, 16-31 = image B).
// Amplitude index = lane & 15; wire w <-> bit (3-w), mask m = 8 >> w.
// All exchanges use xor masks {1,2,4,8} which stay within each 16-lane half.
// ---------------------------------------------------------------------------
__device__ __forceinline__ void lane_ry(float& re, float& im, int amp, int m,
                                        float c, float s) {
  const float pre = __shfl_xor(re, m, 32);
  const float pim = __shfl_xor(im, m, 32);
  if (amp & m) { re = s * pre + c * re;  im = s * pim + c * im; }
  else         { re = c * re - s * pre;  im = c * im - s * pim; }
}

__device__ __forceinline__ void lane_rz(float& re, float& im, int amp, int m,
                                        float c, float s) {
  float r = re, q = im;
  if (amp & m) { re = c * r - s * q;  im = c * q + s * r; }   // *(c + i s)
  else         { re = c * r + s * q;  im = c * q - s * r; }   // *(c - i s)
}

__device__ __forceinline__ void lane_cnot(float& re, float& im, int amp,
                                          int mc, int mt) {
  const float pre = __shfl_xor(re, mt, 32);
  const float pim = __shfl_xor(im, mt, 32);
  if (amp & mc) { re = pre; im = pim; }
}

__device__ __forceinline__ float group16_sum(float v) {
#pragma unroll
  for (int off = 1; off < 16; off <<= 1) v += __shfl_xor(v, off, 32);
  return v;
}

__global__ __launch_bounds__(256) void quanv_kernel(const float* __restrict__ x,
                                                    const float* __restrict__ qp,
                                                    float* __restrict__ featsT /* (784, B) */) {
  const int lane = threadIdx.x & 31;
  const int gwave = (int)((blockIdx.x * blockDim.x + threadIdx.x) >> 5);
  const int amp = lane & 15;                 // amplitude index 0..15
  const int b = gwave * 2 + (lane >> 4);     // image index

  // parameter half-angle trig (16 precise sincos, redundant per lane)
  float cry[8], sry[8], crz[8], srz[8];
#pragma unroll
  for (int g = 0; g < 8; ++g) {
    sincosf(0.5f * qp[2 * g + 0], &sry[g], &cry[g]);
    sincosf(0.5f * qp[2 * g + 1], &srz[g], &crz[g]);
  }

  float re = (amp == 0) ? 1.f : 0.f;         // |0000>
  float im = 0.f;
  const float* xr = x + (size_t)b * 784;

#pragma unroll 1
  for (int p = 0; p < 196; ++p) {
    const int i14 = p / 14, j14 = p % 14;
    const int base = (2 * i14) * 28 + 2 * j14;
    const float v00 = xr[base], v01 = xr[base + 1];
    const float v10 = xr[base + 28], v11 = xr[base + 29];

    float sn, cs;
    __sincosf(0.5f * v00, &sn, &cs); lane_ry(re, im, amp, 8, cs, sn);
    __sincosf(0.5f * v01, &sn, &cs); lane_ry(re, im, amp, 4, cs, sn);
    __sincosf(0.5f * v10, &sn, &cs); lane_ry(re, im, amp, 2, cs, sn);
    __sincosf(0.5f * v11, &sn, &cs); lane_ry(re, im, amp, 1, cs, sn);

#pragma unroll
    for (int l = 0; l < 2; ++l) {
      const int g0 = l * 4;
      lane_ry(re, im, amp, 8, cry[g0 + 0], sry[g0 + 0]); lane_rz(re, im, amp, 8, crz[g0 + 0], srz[g0 + 0]);
      lane_ry(re, im, amp, 4, cry[g0 + 1], sry[g0 + 1]); lane_rz(re, im, amp, 4, crz[g0 + 1], srz[g0 + 1]);
      lane_ry(re, im, amp, 2, cry[g0 + 2], sry[g0 + 2]); lane_rz(re, im, amp, 2, crz[g0 + 2], srz[g0 + 2]);
      lane_ry(re, im, amp, 1, cry[g0 + 3], sry[g0 + 3]); lane_rz(re, im, amp, 1, crz[g0 + 3], srz[g0 + 3]);
      lane_cnot(re, im, amp, 8, 4);   // ctrl 0, tgt 1
      lane_cnot(re, im, amp, 4, 2);   // ctrl 1, tgt 2
      lane_cnot(re, im, amp, 2, 1);   // ctrl 2, tgt 3
    }

    // <Z_w> = sum over amps of (+/-) |amp|^2 ; butterfly over the 16-lane group
    const float pr = re * re + im * im;
    const float e0 = group16_sum((amp & 8) ? -pr : pr);
    const float e1 = group16_sum((amp & 4) ? -pr : pr);
    const float e2 = group16_sum((amp & 2) ? -pr : pr);
    const float e3 = group16_sum((amp & 1) ? -pr : pr);
    if (amp < 4) {
      const float ew = (amp == 0) ? e0 : (amp == 1) ? e1 : (amp == 2) ? e2 : e3;
      featsT[(size_t)(4 * p + amp) * BSZ + b] = ew;
    }
  }
}

// ---------------------------------------------------------------------------
// WMMA fp32 GEMM:  Y(M,N) = At(K,M)^T * Wm(N,K)^T + bias   (i.e. A @ W.T)
// Each wave computes one 16(M) x 64(N) strip with V_WMMA_F32_16X16X4_F32.
// A-frag: lane(0-15)=row M, VGPR0/1 = K+0/K+1; lanes 16-31 get K+2/K+3.
// ---------------------------------------------------------------------------
__global__ __launch_bounds__(256) void gemm_tn_wmma(const float* __restrict__ At,
                                                    const float* __restrict__ Wm,
                                                    const float* __restrict__ bias,
                                                    float* __restrict__ Y,
                                                    int M, int N, int K) {
  const int lane = threadIdx.x & 31;
  const int wave = threadIdx.x >> 5;
  const int gw = blockIdx.x * 8 + wave;
  const int nStrips = N >> 6;
  const int m0 = (gw / nStrips) * 16;
  const int n0 = (gw % nStrips) * 64;
  const int row = lane & 15;
  const int kh = (lane >> 4) << 1;   // 0 or 2

  const float* ap  = At + m0 + row;
  const float* wp0 = Wm + (size_t)(n0 +  0 + row) * K;
  const float* wp1 = Wm + (size_t)(n0 + 16 + row) * K;
  const float* wp2 = Wm + (size_t)(n0 + 32 + row) * K;
  const float* wp3 = Wm + (size_t)(n0 + 48 + row) * K;

  v8f acc0 = {0.f,0.f,0.f,0.f,0.f,0.f,0.f,0.f};
  v8f acc1 = acc0, acc2 = acc0, acc3 = acc0;

#pragma unroll 2
  for (int k0 = 0; k0 < K; k0 += 4) {
    const int ka = k0 + kh;
    v2f a;  a.x  = ap[(size_t)ka * M];  a.y  = ap[(size_t)(ka + 1) * M];
    v2f b0; b0.x = wp0[ka];             b0.y = wp0[ka + 1];
    v2f b1; b1.x = wp1[ka];             b1.y = wp1[ka + 1];
    v2f b2; b2.x = wp2[ka];             b2.y = wp2[ka + 1];
    v2f b3; b3.x = wp3[ka];             b3.y = wp3[ka + 1];
    acc0 = __builtin_amdgcn_wmma_f32_16x16x4_f32(false, a, false, b0, (short)0, acc0, false, false);
    acc1 = __builtin_amdgcn_wmma_f32_16x16x4_f32(false, a, false, b1, (short)0, acc1, false, false);
    acc2 = __builtin_amdgcn_wmma_f32_16x16x4_f32(false, a, false, b2, (short)0, acc2, false, false);
    acc3 = __builtin_amdgcn_wmma_f32_16x16x4_f32(false, a, false, b3, (short)0, acc3, false, false);
  }

  // C layout: lane (L&15) = N column, VGPR r -> M = r + 8*(L>>4)
  const int mo = m0 + ((lane >> 4) << 3);
  const float bi0 = bias[n0 + row];
  const float bi1 = bias[n0 + 16 + row];
  const float bi2 = bias[n0 + 32 + row];
  const float bi3 = bias[n0 + 48 + row];
#pragma unroll
  for (int r = 0; r < 8; ++r) {
    float* yr = Y + (size_t)(mo + r) * N + n0 + row;
    yr[0]  = acc0[r] + bi0;
    yr[16] = acc1[r] + bi1;
    yr[32] = acc2[r] + bi2;
    yr[48] = acc3[r] + bi3;
  }
}

// ---------------------------------------------------------------------------
// BatchNorm (training mode, biased stats) helpers
// ---------------------------------------------------------------------------
__global__ void zero_kernel(float* __restrict__ p, int n) {
  int i = blockIdx.x * blockDim.x + threadIdx.x;
  if (i < n) p[i] = 0.f;
}

__global__ __launch_bounds__(256) void colstats_kernel(const float* __restrict__ Y,
                                                       float* __restrict__ sum,
                                                       float* __restrict__ sumsq) {
  const int j = threadIdx.x;            // column 0..255 (coalesced reads)
  const size_t r0 = (size_t)blockIdx.x * 128;
  float s = 0.f, s2 = 0.f;
#pragma unroll 4
  for (int r = 0; r < 128; ++r) {
    float v = Y[(r0 + r) * NH + j];
    s += v; s2 += v * v;
  }
  atomicAdd(&sum[j], s);
  atomicAdd(&sumsq[j], s2);
}

__global__ void finalize_stats_kernel(const float* __restrict__ sum,
                                      const float* __restrict__ sumsq,
                                      const float* __restrict__ g,
                                      const float* __restrict__ be,
                                      float* __restrict__ scale,
                                      float* __restrict__ shift) {
  const int j = threadIdx.x;
  const float inv = 1.f / (float)BSZ;
  float mean = sum[j] * inv;
  float var  = sumsq[j] * inv - mean * mean;
  float rstd = rsqrtf(var + 1e-5f);
  float sc = g[j] * rstd;
  scale[j] = sc;
  shift[j] = be[j] - mean * sc;
}

__global__ __launch_bounds__(256) void bnrelu_t_kernel(const float* __restrict__ Y,
                                                       const float* __restrict__ scale,
                                                       const float* __restrict__ shift,
                                                       float* __restrict__ hT /* (256,B) */) {
  const size_t idx = (size_t)blockIdx.x * blockDim.x + threadIdx.x; // B*256
  const int j = (int)(idx & 255);
  const size_t b = idx >> 8;
  float v = fmaxf(scale[j] * Y[idx] + shift[j], 0.f);
  hT[(size_t)j * BSZ + b] = v;
}

__global__ __launch_bounds__(256) void bnrelu_add_kernel(const float* __restrict__ Y,
                                                         const float* __restrict__ scale,
                                                         const float* __restrict__ shift,
                                                         const float* __restrict__ hT,
                                                         float* __restrict__ h2 /* (B,256) */) {
  const size_t idx = (size_t)blockIdx.x * blockDim.x + threadIdx.x;
  const int j = (int)(idx & 255);
  const size_t b = idx >> 8;
  h2[idx] = fmaxf(scale[j] * Y[idx] + shift[j], 0.f) + hT[(size_t)j * BSZ + b];
}

// ---------------------------------------------------------------------------
// Head: logits = h2 @ w3.T + b3 then log_softmax. One wave32 per row.
// ---------------------------------------------------------------------------
__global__ __launch_bounds__(256) void head_kernel(const float* __restrict__ h2,
                                                   const float* __restrict__ w3,
                                                   const float* __restrict__ b3,
                                                   float* __restrict__ out) {
  const int rowIdx = (int)((blockIdx.x * blockDim.x + threadIdx.x) >> 5);
  const int lane = threadIdx.x & 31;
  if (rowIdx >= BSZ) return;
  const float* hp = h2 + (size_t)rowIdx * NH;

  float acc[10];
#pragma unroll
  for (int n = 0; n < 10; ++n) acc[n] = 0.f;
#pragma unroll
  for (int i = 0; i < 8; ++i) {
    const int k = lane + 32 * i;
    const float hv = hp[k];
#pragma unroll
    for (int n = 0; n < 10; ++n) acc[n] += hv * w3[n * NH + k];
  }
#pragma unroll
  for (int n = 0; n < 10; ++n)
    for (int off = 16; off > 0; off >>= 1)
      acc[n] += __shfl_xor(acc[n], off, 32);

  float logit[10], mx = -INFINITY;
#pragma unroll
  for (int n = 0; n < 10; ++n) { logit[n] = acc[n] + b3[n]; mx = fmaxf(mx, logit[n]); }
  float se = 0.f;
#pragma unroll
  for (int n = 0; n < 10; ++n) se += expf(logit[n] - mx);
  const float lse = logf(se) + mx;
  if (lane < 10) out[(size_t)rowIdx * 10 + lane] = logit[lane] - lse;
}

// ---------------------------------------------------------------------------
extern "C" void kernel_launch(void* const* d_in, const int* in_sizes, int n_in,
                              void* d_out, int out_size, void* d_ws, size_t ws_size,
                              hipStream_t stream) {
  const float* x   = (const float*)d_in[0];
  const float* qp  = (const float*)d_in[1];
  const float* w1  = (const float*)d_in[2];
  const float* b1  = (const float*)d_in[3];
  const float* g1  = (const float*)d_in[4];
  const float* be1 = (const float*)d_in[5];
  const float* w2  = (const float*)d_in[6];
  const float* b2  = (const float*)d_in[7];
  const float* g2  = (const float*)d_in[8];
  const float* be2 = (const float*)d_in[9];
  const float* w3  = (const float*)d_in[10];
  const float* b3  = (const float*)d_in[11];
  float* out = (float*)d_out;

  float* W = (float*)d_ws;
  float* featsT = W;                                  // 784*B floats (51.4 MB)
  float* hT     = W;                                  // 256*B, reuses featsT region after GEMM1
  float* Y      = W + (size_t)NFEAT * BSZ;            // B*256
  float* h2     = Y + (size_t)BSZ * NH;               // B*256
  float* stats  = h2 + (size_t)BSZ * NH;              // 1024 floats
  float* sum    = stats;
  float* sumsq  = stats + 256;
  float* scale  = stats + 512;
  float* shift  = stats + 768;

  const int gemmBlocks = (BSZ / 16) * (NH / 64) / 8;  // 512 blocks of 8 waves

  // 16 lanes per image, 2 images per wave: BSZ/2 waves = BSZ*16 threads
  quanv_kernel<<<(BSZ * 16) / 256, 256, 0, stream>>>(x, qp, featsT);

  gemm_tn_wmma<<<gemmBlocks, 256, 0, stream>>>(featsT, w1, b1, Y, BSZ, NH, NFEAT);
  zero_kernel<<<2, 256, 0, stream>>>(sum, 512);
  colstats_kernel<<<BSZ / 128, 256, 0, stream>>>(Y, sum, sumsq);
  finalize_stats_kernel<<<1, 256, 0, stream>>>(sum, sumsq, g1, be1, scale, shift);
  bnrelu_t_kernel<<<(BSZ * NH) / 256, 256, 0, stream>>>(Y, scale, shift, hT);

  gemm_tn_wmma<<<gemmBlocks, 256, 0, stream>>>(hT, w2, b2, Y, BSZ, NH, NH);
  zero_kernel<<<2, 256, 0, stream>>>(sum, 512);
  colstats_kernel<<<BSZ / 128, 256, 0, stream>>>(Y, sum, sumsq);
  finalize_stats_kernel<<<1, 256, 0, stream>>>(sum, sumsq, g2, be2, scale, shift);
  bnrelu_add_kernel<<<(BSZ * NH) / 256, 256, 0, stream>>>(Y, scale, shift, hT, h2);

  head_kernel<<<(BSZ * 32) / 256, 256, 0, stream>>>(h2, w3, b3, out);
}